// GCNConvSC_38319698215460
// MI455X (gfx1250) — compile-verified
//
#include <hip/hip_runtime.h>

typedef float v8f __attribute__((ext_vector_type(8)));
typedef float v2f __attribute__((ext_vector_type(2)));

#define FEAT 128

// ---------------------------------------------------------------- degree ----
__global__ void k_deg_init(float* __restrict__ deg, int N) {
    int i = blockIdx.x * blockDim.x + threadIdx.x;
    if (i < N) deg[i] = 1.0f;  // self loop contributes 1 to every node
}

__global__ void k_deg_count(const int* __restrict__ ei, float* __restrict__ deg, int E) {
    int e = blockIdx.x * blockDim.x + threadIdx.x;
    if (e < E) atomicAdd(&deg[ei[E + e]], 1.0f);  // degree counted at dst
}

__global__ void k_dinv(float* __restrict__ deg, int N) {
    int i = blockIdx.x * blockDim.x + threadIdx.x;
    if (i < N) deg[i] = rsqrtf(deg[i]);  // deg >= 1 always (self loop)
}

// ----------------------------------------------- self-loop init of acc ------
// acc[i,:] = dinv[i]^2 * x[i,:]   (self-loop message, norm = dinv*dinv)
__global__ void k_acc_init(const float* __restrict__ x, const float* __restrict__ dinv,
                           float* __restrict__ acc, int N) {
    int i = blockIdx.x * blockDim.x + threadIdx.x;  // over N*32 float4 groups
    int node = i >> 5;
    if (node < N) {
        float di = dinv[node];
        float s  = di * di;
        float4 xv = ((const float4*)x)[i];
        float4 o;
        o.x = s * xv.x; o.y = s * xv.y; o.z = s * xv.z; o.w = s * xv.w;
        ((float4*)acc)[i] = o;
    }
}

// ----------------------------------------------------------- edge scatter ---
// acc[dst,:] += dinv[src]*dinv[dst] * x[src,:]
// 32 threads (one wave-uniform edge) cover one edge's 128 feats as float4.
// Edge index is wave-uniform: readfirstlane makes it explicit so the index
// and dinv loads scalarize (s_load), leaving only the coalesced float4
// gather (L2-resident: x = 51.2MB < 192MB L2) and 4 f32 atomics per thread.
__global__ void k_scatter(const float* __restrict__ x, const float* __restrict__ dinv,
                          const int* __restrict__ ei, float* __restrict__ acc, int E) {
    long long idx = (long long)blockIdx.x * blockDim.x + threadIdx.x;
    int e = (int)(idx >> 5);
    if (e >= E) return;
    e = __builtin_amdgcn_readfirstlane(e);  // wave-uniform by construction
    int q = (int)(idx & 31);
    int s = ei[e];
    int d = ei[E + e];
    float nrm = dinv[s] * dinv[d];
    float4 xv = ((const float4*)(x + (size_t)s * FEAT))[q];
    float* ap = acc + (size_t)d * FEAT + q * 4;
    atomicAdd(ap + 0, nrm * xv.x);
    atomicAdd(ap + 1, nrm * xv.y);
    atomicAdd(ap + 2, nrm * xv.z);
    atomicAdd(ap + 3, nrm * xv.w);
}

// ------------------------------------------------------- WMMA f32 GEMM ------
// out = x + acc @ W + b  using V_WMMA_F32_16X16X4_F32.
// Block = 256 threads = 8 waves; each wave computes a 16-row tile across all
// 128 output columns (8 accumulator tiles of v8f).
//
// W is staged in LDS *transposed + XOR-swizzled*:
//     sW[n*128 + (k ^ ((n&15)<<2))] = W[k][n]
//  - (k,k+1) stay contiguous (xor value is a multiple of 4, k is even)
//    -> each B fragment is ONE ds_load_b64, no register packing movs
//  - banks k0^4*nlo (lanes 0-15) vs k0^2^4*nlo (lanes 16-31): 64 distinct
//    bank-dwords per wave-wide b64 access -> conflict-free
//  - swizzled k-offset is independent of the n-tile index t, so all 8 B
//    loads share one base VGPR with immediate offsets t*8192 bytes
__global__ __launch_bounds__(256) void k_gemm_wmma(
    const float* __restrict__ acc, const float* __restrict__ x,
    const float* __restrict__ W, const float* __restrict__ b,
    float* __restrict__ out, int N)
{
    __shared__ float sW[FEAT * FEAT];  // 64 KB
    const int tid = threadIdx.x;
    for (int i = tid; i < FEAT * FEAT / 4; i += 256) {
        float4 v = ((const float4*)W)[i];
        int k  = i >> 5;
        int n0 = (i & 31) << 2;
        float vv[4] = {v.x, v.y, v.z, v.w};
        #pragma unroll
        for (int j = 0; j < 4; ++j) {
            int n = n0 + j;
            sW[n * FEAT + (k ^ ((n & 15) << 2))] = vv[j];
        }
    }
    __syncthreads();

    const int wave  = tid >> 5;
    const int lane  = tid & 31;
    const int m0    = (blockIdx.x * 8 + wave) * 16;       // row-tile base
    const int mrow  = m0 + (lane & 15);
    const int mload = (mrow < N) ? mrow : (N - 1);        // clamp OOB loads
    const int khalf = (lane >> 4) << 1;                   // lanes 0-15: K 0,1; 16-31: K 2,3
    const int nlo   = lane & 15;

    v8f c[8];
    #pragma unroll
    for (int t = 0; t < 8; ++t) c[t] = (v8f){0, 0, 0, 0, 0, 0, 0, 0};

    for (int k0 = 0; k0 < FEAT; k0 += 4) {
        // A fragment: 16x4 f32, lane holds acc[m][k0+khalf], acc[m][k0+khalf+1]
        const float* apx = acc + (size_t)mload * FEAT + (k0 + khalf);
        v2f a;
        a.x = apx[0];
        a.y = apx[1];
        // B fragments: swizzled k-offset shared by all 8 n-tiles
        const int kx = (k0 + khalf) ^ (nlo << 2);
        const float* bp = &sW[nlo * FEAT + kx];
        #pragma unroll
        for (int t = 0; t < 8; ++t) {
            v2f bb = *(const v2f*)(bp + t * 2048);  // ds_load_b64, offset t*8192B
            c[t] = __builtin_amdgcn_wmma_f32_16x16x4_f32(
                false, a, false, bb, (short)0, c[t], false, false);
        }
    }

    // C/D layout: VGPR j -> M = m0 + j + (lane>=16 ? 8 : 0), N = tile*16 + (lane&15)
    const int rbase = m0 + ((lane >> 4) << 3);
    #pragma unroll
    for (int t = 0; t < 8; ++t) {
        const int col    = t * 16 + nlo;
        const float bias = b[col];
        #pragma unroll
        for (int j = 0; j < 8; ++j) {
            const int r = rbase + j;
            if (r < N)
                out[(size_t)r * FEAT + col] = x[(size_t)r * FEAT + col] + c[t][j] + bias;
        }
    }
}

// ---------------------------------------------------------------------------
extern "C" void kernel_launch(void* const* d_in, const int* in_sizes, int n_in,
                              void* d_out, int out_size, void* d_ws, size_t ws_size,
                              hipStream_t stream) {
    const float* x  = (const float*)d_in[0];
    const int*   ei = (const int*)d_in[1];   // [2, E] flat: [0..E) = src, [E..2E) = dst
    const float* W  = (const float*)d_in[2];
    const float* b  = (const float*)d_in[3];
    float* out = (float*)d_out;

    const int N = in_sizes[0] / FEAT;
    const int E = in_sizes[1] / 2;

    // Workspace layout: deg/dinv [N floats], then acc [N*128 floats] (~51.6 MB)
    float* deg = (float*)d_ws;
    size_t accOff = (((size_t)N + 1023) / 1024) * 1024;
    float* acc = deg + accOff;

    const int T = 256;
    k_deg_init <<<(N + T - 1) / T, T, 0, stream>>>(deg, N);
    k_deg_count<<<(E + T - 1) / T, T, 0, stream>>>(ei, deg, E);
    k_dinv     <<<(N + T - 1) / T, T, 0, stream>>>(deg, N);

    long long g1 = (long long)N * 32;  // N*128/4 float4 chunks
    k_acc_init <<<(int)((g1 + T - 1) / T), T, 0, stream>>>(x, deg, acc, N);

    long long g2 = (long long)E * 32;  // E * 128/4 float4 chunks
    k_scatter  <<<(int)((g2 + T - 1) / T), T, 0, stream>>>(x, deg, ei, acc, E);

    int gb = (N + 127) / 128;  // 8 waves/block * 16 rows/wave
    k_gemm_wmma<<<gb, 256, 0, stream>>>(acc, x, W, b, out, N);
}